// TGQN_37830071943787
// MI455X (gfx1250) — compile-verified
//
#include <hip/hip_runtime.h>
#include <hip/hip_bf16.h>
#include <math.h>

// ---------------- problem constants (match reference) ----------------
#define B_   1024
#define S_   20
#define D_   128
#define E_   128
#define X_   136   // D + R
#define NI_  100000

typedef __attribute__((ext_vector_type(2))) float v2f;
typedef __attribute__((ext_vector_type(8))) float v8f;
typedef __attribute__((ext_vector_type(4))) unsigned int u32x4;
typedef __attribute__((ext_vector_type(4))) int i32x4;
typedef __attribute__((ext_vector_type(8))) int i32x8;

#if defined(__has_builtin)
#if __has_builtin(__builtin_amdgcn_tensor_load_to_lds) && __has_builtin(__builtin_amdgcn_s_wait_tensorcnt)
#define HAVE_TDM 1
#endif
#endif

__device__ __forceinline__ float sigmoid_f(float v) {
    return 1.0f / (1.0f + __expf(-v));
}

#ifdef HAVE_TDM
// -------- Tensor Data Mover helpers (D# per CDNA5 ISA) --
__device__ __forceinline__ void tdm_issue(u32x4 g0, i32x8 g1, i32x4 g2, i32x4 g3) {
#if __clang_major__ >= 23
    i32x8 z8 = {0, 0, 0, 0, 0, 0, 0, 0};
    __builtin_amdgcn_tensor_load_to_lds(g0, g1, g2, g3, z8, 0);
#else
    __builtin_amdgcn_tensor_load_to_lds(g0, g1, g2, g3, 0);
#endif
}

// Copy `rows` rows of `row_len` fp32 (row-contiguous tensor) into LDS at byte
// offset lds_off.  valid_rows = tensor_dim1 so OOB rows are zero-filled by the
// TDM.  pad_dw>0 inserts pad_dw DWORDs of LDS padding after each 128-DWORD
// interval (our bank-conflict row padding, done by the DMA engine).
__device__ __forceinline__ void tdm_load_rows(const float* gbase, unsigned lds_off,
                                              unsigned rows, unsigned valid_rows,
                                              unsigned row_len, int pad_dw) {
    unsigned long long ga = (unsigned long long)(size_t)gbase;
    u32x4 g0;
    g0[0] = 1u;                                              // count=1 user D#
    g0[1] = lds_off;                                         // lds_addr (bytes)
    g0[2] = (unsigned)(ga & 0xFFFFFFFFu);                    // global_addr[31:0]
    g0[3] = (unsigned)((ga >> 32) & 0x1FFFFFFu) | (2u << 30);// ga[56:32] | type=2
    int w0 = (2 << 16);                                      // data_size = 4B
    if (pad_dw > 0)
        w0 |= (1 << 20) | (6 << 22) | ((pad_dw - 1) << 25);  // pad_en|intv=128dw|amount
    i32x8 g1;
    g1[0] = w0;
    g1[1] = (int)((row_len & 0xFFFFu) << 16);                // tensor_dim0 lo16
    g1[2] = (int)(((row_len >> 16) & 0xFFFFu) |              // tensor_dim0 hi16
                  ((valid_rows & 0xFFFFu) << 16));           // tensor_dim1 lo16
    g1[3] = (int)(((valid_rows >> 16) & 0xFFFFu) |           // tensor_dim1 hi16
                  ((row_len & 0xFFFFu) << 16));              // tile_dim0
    g1[4] = (int)(rows & 0xFFFFu);                           // tile_dim1 (tile_dim2=0)
    g1[5] = (int)row_len;                                    // tensor_dim0_stride
    g1[6] = 0;
    g1[7] = 0;
    i32x4 z4 = {0, 0, 0, 0};
    tdm_issue(g0, g1, z4, z4);
}

// Gather-mode DMA: fetch 8 rows of 128 fp32 selected by 32-bit row indices
// (groups 2/3 carry the indices), LDS row stride = 128 + pad_dw DWORDs.
__device__ __forceinline__ void tdm_gather8(const float* gbase, unsigned lds_off,
                                            const int* idx8, int pad_dw) {
    unsigned long long ga = (unsigned long long)(size_t)gbase;
    u32x4 g0;
    g0[0] = 1u | (1u << 30) | (1u << 31);        // count=1 | 32-bit idx | gather_mode
    g0[1] = lds_off;
    g0[2] = (unsigned)(ga & 0xFFFFFFFFu);
    g0[3] = (unsigned)((ga >> 32) & 0x1FFFFFFu) | (2u << 30);
    int w0 = (2 << 16);
    if (pad_dw > 0)
        w0 |= (1 << 20) | (6 << 22) | ((pad_dw - 1) << 25);
    i32x8 g1;
    g1[0] = w0;
    g1[1] = (int)(128u << 16);                   // tensor_dim0 = 128
    g1[2] = (int)((NI_ & 0xFFFF) << 16);         // tensor_dim1 = NI lo16
    g1[3] = (int)(((NI_ >> 16) & 0xFFFF) | (128u << 16)); // dim1 hi16 | tile_dim0=128
    g1[4] = 8;                                   // tile_dim1 = #valid indices
    g1[5] = 128;                                 // tensor_dim0_stride = 128
    g1[6] = 0;
    g1[7] = 0;
    i32x4 g2, g3;
    g2[0] = idx8[0]; g2[1] = idx8[1]; g2[2] = idx8[2]; g2[3] = idx8[3];
    g3[0] = idx8[4]; g3[1] = idx8[5]; g3[2] = idx8[6]; g3[3] = idx8[7];
    tdm_issue(g0, g1, g2, g3);
}
#endif // HAVE_TDM

// ---------------------------------------------------------------------
// Kernel 0: zero the segment-sum / count scratch.
// ---------------------------------------------------------------------
__global__ __launch_bounds__(256) void k_zero(float* __restrict__ sums,
                                              float* __restrict__ counts) {
    const long long total_sums = (long long)NI_ * D_;
    const long long total = total_sums + NI_;
    long long i = (long long)blockIdx.x * 256 + threadIdx.x;
    if (i < total_sums) sums[i] = 0.0f;
    else if (i < total)  counts[i - total_sums] = 0.0f;
}

// ---------------------------------------------------------------------
// Kernel 1: fused gather + GRU + segment-sum scatter.
// 128 threads handle 8 users; thread t owns hidden unit t.  Per step the
// 8 item rows are fetched with one TDM gather-mode descriptor (32-bit
// indices, pad_amount=8 DW reproducing the xs[8][136] stride); h_new is
// scattered straight into item_sums/counts (seq_h never materialized).
// Dynamic LDS only, so descriptor lds_addr offsets are exact.
// ---------------------------------------------------------------------
__global__ __launch_bounds__(128) void k_gru(
    const int*   __restrict__ users_items,   // (B,S)
    const float* __restrict__ users_rewards, // (B,S)
    const float* __restrict__ items_table,   // (NI,D)
    const float* __restrict__ Wx,            // (384,136)
    const float* __restrict__ Wh,            // (384,128)
    const float* __restrict__ bx,            // (384,)
    const float* __restrict__ bh,            // (384,)
    float* __restrict__ hT,                  // (B,D) out
    float* __restrict__ item_sums,           // (NI,D) accum
    float* __restrict__ item_counts)         // (NI,) accum
{
    extern __shared__ float gsm[];
    float (*xs)[X_] = (float(*)[X_])gsm;                 // 8 x 136, offset 0
    float (*hs)[D_] = (float(*)[D_])(gsm + 8 * X_);      // 8 x 128
    int*   item_s   = (int*)(gsm + 8 * X_ + 8 * D_);     // 8 ids

    const int t  = threadIdx.x;
    const int b0 = blockIdx.x * 8;

    float hreg[8];
    #pragma unroll
    for (int u = 0; u < 8; ++u) { hreg[u] = 0.0f; hs[u][t] = 0.0f; }

    const float bxr = bx[t], bxz = bx[t + 128], bxn = bx[t + 256];
    const float bhr = bh[t], bhz = bh[t + 128], bhn = bh[t + 256];

    const float* wxr = Wx + (size_t)t * X_;
    const float* wxz = Wx + (size_t)(t + 128) * X_;
    const float* wxn = Wx + (size_t)(t + 256) * X_;
    const float* whr = Wh + (size_t)t * D_;
    const float* whz = Wh + (size_t)(t + 128) * D_;
    const float* whn = Wh + (size_t)(t + 256) * D_;

    for (int s = 0; s < S_; ++s) {
        if (t < 8) item_s[t] = users_items[(size_t)(b0 + t) * S_ + s];
        __syncthreads();   // ids visible; prev-step hs/xs reads complete

#ifdef HAVE_TDM
        if (t < 32) {
            int ids[8];
            #pragma unroll
            for (int u = 0; u < 8; ++u)      // uniform addresses -> scalar loads
                ids[u] = users_items[(size_t)(b0 + u) * S_ + s];
            tdm_gather8(items_table, /*lds_off=*/0, ids, /*pad_dw=*/8);
        }
#else
        #pragma unroll
        for (int u = 0; u < 8; ++u)
            xs[u][t] = items_table[(size_t)item_s[u] * D_ + t];
#endif
        if (t < 64) {
            int u = t >> 3, j = t & 7;       // pad region untouched by TDM
            xs[u][D_ + j] = users_rewards[(size_t)(b0 + u) * S_ + s];
        }
#ifdef HAVE_TDM
        if (t < 32) __builtin_amdgcn_s_wait_tensorcnt(0);
#endif
        __syncthreads();

        float accr[8], accz[8], gxn[8], ghn[8];
        #pragma unroll
        for (int u = 0; u < 8; ++u) {
            accr[u] = bxr + bhr; accz[u] = bxz + bhz;
            gxn[u] = bxn;        ghn[u] = bhn;
        }
        for (int k = 0; k < X_; ++k) {
            float wr = wxr[k], wz = wxz[k], wn = wxn[k];
            #pragma unroll
            for (int u = 0; u < 8; ++u) {
                float xv = xs[u][k];
                accr[u] = fmaf(wr, xv, accr[u]);
                accz[u] = fmaf(wz, xv, accz[u]);
                gxn[u]  = fmaf(wn, xv, gxn[u]);
            }
        }
        for (int k = 0; k < D_; ++k) {
            float wr = whr[k], wz = whz[k], wn = whn[k];
            #pragma unroll
            for (int u = 0; u < 8; ++u) {
                float hv = hs[u][k];
                accr[u] = fmaf(wr, hv, accr[u]);
                accz[u] = fmaf(wz, hv, accz[u]);
                ghn[u]  = fmaf(wn, hv, ghn[u]);
            }
        }
        __syncthreads();   // done reading hs/xs before overwrite

        #pragma unroll
        for (int u = 0; u < 8; ++u) {
            float r  = sigmoid_f(accr[u]);
            float z  = sigmoid_f(accz[u]);
            float n  = tanhf(gxn[u] + r * ghn[u]);
            float hn = (1.0f - z) * n + z * hreg[u];
            hreg[u] = hn;
            hs[u][t] = hn;
            atomicAdd(&item_sums[(size_t)item_s[u] * D_ + t], hn);
        }
        if (t < 8) atomicAdd(&item_counts[item_s[t]], 1.0f);
        __syncthreads();
    }

    #pragma unroll
    for (int u = 0; u < 8; ++u)
        hT[(size_t)(b0 + u) * D_ + t] = hreg[u];
}

// ---------------------------------------------------------------------
// Kernel 2: users_emb[b] = hT[winner(ids[b])] @ W_u.
// scatter-set with duplicate indices -> last index wins, then gather.
// ---------------------------------------------------------------------
__global__ __launch_bounds__(128) void k_users_emb(
    const int*   __restrict__ users_ids,
    const float* __restrict__ hT,
    const float* __restrict__ W_u,          // (D,E) row-major
    float* __restrict__ users_emb)          // (B,E)
{
    const int b = blockIdx.x, t = threadIdx.x;
    __shared__ int   red[128];
    __shared__ float h[D_];

    const int myid = users_ids[b];
    int best = -1;
    for (int j = t; j < B_; j += 128)
        if (users_ids[j] == myid) best = max(best, j);
    red[t] = best;
    __syncthreads();
    for (int off = 64; off > 0; off >>= 1) {
        if (t < off) red[t] = max(red[t], red[t + off]);
        __syncthreads();
    }
    const int bwin = red[0];

    h[t] = hT[(size_t)bwin * D_ + t];
    __syncthreads();

    float acc = 0.0f;
    for (int k = 0; k < D_; ++k)
        acc = fmaf(h[k], W_u[(size_t)k * E_ + t], acc);
    users_emb[(size_t)b * E_ + t] = acc;
}

// ---------------------------------------------------------------------
// Kernel 3: items_emb = items_input @ W_i with V_WMMA_F32_16X16X4_F32.
// W_i (64KB) is TDM-copied to LDS (overlapped with the fused EMA tile
// build); items_input rows are built on the fly into padded LDS.
// 97KB dynamic LDS -> 3 workgroups/WGP on CDNA5's 320KB LDS.
// ---------------------------------------------------------------------
__global__ __launch_bounds__(128) void k_items_emb(
    const float* __restrict__ items_memory,  // (NI,D)
    const float* __restrict__ items_table,   // (NI,D)
    const float* __restrict__ item_sums,     // (NI,D)
    const float* __restrict__ item_counts,   // (NI,)
    const float* __restrict__ W_i,           // (D,E) row-major
    float* __restrict__ items_emb)           // (NI,E)
{
    extern __shared__ float ism[];
    float* wi = ism;                                       // 128x128, offset 0
    float (*itile)[D_ + 2] = (float(*)[D_ + 2])(ism + D_ * E_);  // 64 x 130

    const int t = threadIdx.x, lane = t & 31, w = t >> 5;
    const int i0 = blockIdx.x * 64;

#ifdef HAVE_TDM
    if (t < 32)   // one contiguous 16384-dword copy, no padding
        tdm_load_rows(W_i, /*lds_off=*/0, 1, 1, D_ * E_, 0);
#else
    for (int idx = t; idx < D_ * E_; idx += 128) wi[idx] = W_i[idx];
#endif

    // items_input = (touched ? 0.5*mem + 0.5*sums/cnt : mem) + table
    for (int r = 0; r < 64; ++r) {
        const int i = i0 + r;
        float v = 0.0f;
        if (i < NI_) {
            float cnt = item_counts[i];
            float mem = items_memory[(size_t)i * D_ + t];
            float tab = items_table[(size_t)i * D_ + t];
            if (cnt > 0.0f) {
                float mean = item_sums[(size_t)i * D_ + t] / cnt;
                v = 0.5f * mem + 0.5f * mean + tab;
            } else {
                v = mem + tab;
            }
        }
        itile[r][t] = v;
    }
#ifdef HAVE_TDM
    if (t < 32) __builtin_amdgcn_s_wait_tensorcnt(0);
#endif
    __syncthreads();

    const int m    = lane & 15;
    const int koff = (lane < 16) ? 0 : 2;    // f32 fragment K-split per half-wave
    const int row  = w * 16 + m;

    const v8f zero = {0.f,0.f,0.f,0.f,0.f,0.f,0.f,0.f};
    v8f c[8];
    #pragma unroll
    for (int nt = 0; nt < 8; ++nt) c[nt] = zero;

    for (int k0 = 0; k0 < D_; k0 += 4) {
        v2f a;
        a.x = itile[row][k0 + koff];
        a.y = itile[row][k0 + koff + 1];
        #pragma unroll
        for (int nt = 0; nt < 8; ++nt) {
            v2f bf;
            bf.x = wi[(k0 + koff) * E_ + nt * 16 + m];
            bf.y = wi[(k0 + koff + 1) * E_ + nt * 16 + m];
            c[nt] = __builtin_amdgcn_wmma_f32_16x16x4_f32(
                        false, a, false, bf, (short)0, c[nt], false, false);
        }
    }

    #pragma unroll
    for (int nt = 0; nt < 8; ++nt) {
        #pragma unroll
        for (int j = 0; j < 8; ++j) {
            const int mm = j + ((lane < 16) ? 0 : 8);   // C: VGPR j = rows j/j+8
            const int gi = i0 + w * 16 + mm;
            if (gi < NI_)
                items_emb[(size_t)gi * E_ + nt * 16 + m] = c[nt][j];
        }
    }
}

// ---------------------------------------------------------------------
// Kernel 4: out = users_emb @ items_emb^T  (1024 x 100000, fp32 WMMA).
// Each block walks IT_OUT=4 consecutive 48-item tiles with ping-pong LDS
// buffers: TDM prefetches tile it+1 while tile it is computed
// (TENSORcnt overlap).  TDM pad_amount=2 DW gives the conflict-free
// row stride 130; OOB rows past NI are zero-filled by the DMA.
// ---------------------------------------------------------------------
#define TI_OUT 48
#define IT_OUT 4

__global__ __launch_bounds__(128) void k_out(
    const float* __restrict__ users_emb,   // (B,E)
    const float* __restrict__ items_emb,   // (NI,E)
    float* __restrict__ out)               // (B,NI)
{
    extern __shared__ float osm[];
    const int LDW = D_ + 2;                                        // 130
    float (*ua)[D_ + 2]    = (float(*)[D_ + 2])osm;                // 64 rows
    float (*ibuf0)[D_ + 2] = (float(*)[D_ + 2])(osm + 64 * LDW);
    float (*ibuf1)[D_ + 2] = (float(*)[D_ + 2])(osm + (64 + TI_OUT) * LDW);
    const unsigned UA_BYTES = 64u * LDW * 4u;
    const unsigned IB_BYTES = (unsigned)TI_OUT * LDW * 4u;

    const int t = threadIdx.x, lane = t & 31, w = t >> 5;
    const int ibase = blockIdx.x * (TI_OUT * IT_OUT);
    const int u0 = blockIdx.y * 64;

#ifdef HAVE_TDM
    if (t < 32) {
        tdm_load_rows(users_emb + (size_t)u0 * E_, 0, 64, 64, D_, 2);
        {
            int r0 = ibase;
            unsigned rv = (r0 < NI_) ? (unsigned)(NI_ - r0) : 0u;
            tdm_load_rows(items_emb + (size_t)r0 * E_, UA_BYTES, TI_OUT, rv, D_, 2);
        }
        {
            int r0 = ibase + TI_OUT;
            unsigned rv = (r0 < NI_) ? (unsigned)(NI_ - r0) : 0u;
            tdm_load_rows(items_emb + (size_t)r0 * E_, UA_BYTES + IB_BYTES, TI_OUT, rv, D_, 2);
        }
        __builtin_amdgcn_s_wait_tensorcnt(1);   // ua + tile0 done, tile1 in flight
    }
#else
    for (int idx = t; idx < 64 * 32; idx += 128) {
        const int r = idx >> 5, cq = idx & 31;
        float4 v = ((const float4*)(users_emb + (size_t)(u0 + r) * E_))[cq];
        ua[r][cq*4+0] = v.x; ua[r][cq*4+1] = v.y;
        ua[r][cq*4+2] = v.z; ua[r][cq*4+3] = v.w;
    }
#endif
    __syncthreads();

    const int m    = lane & 15;
    const int koff = (lane < 16) ? 0 : 2;

    for (int it = 0; it < IT_OUT; ++it) {
        const int i0 = ibase + it * TI_OUT;
        float (*ib)[D_ + 2] = (it & 1) ? ibuf1 : ibuf0;

#ifndef HAVE_TDM
        for (int idx = t; idx < TI_OUT * 32; idx += 128) {
            const int r = idx >> 5, cq = idx & 31;
            const int gi = i0 + r;
            float4 v = make_float4(0.f, 0.f, 0.f, 0.f);
            if (gi < NI_)
                v = ((const float4*)(items_emb + (size_t)gi * E_))[cq];
            ib[r][cq*4+0] = v.x; ib[r][cq*4+1] = v.y;
            ib[r][cq*4+2] = v.z; ib[r][cq*4+3] = v.w;
        }
        __syncthreads();
#endif

        const v8f zero = {0.f,0.f,0.f,0.f,0.f,0.f,0.f,0.f};
        v8f c[3];
        #pragma unroll
        for (int nt = 0; nt < 3; ++nt) c[nt] = zero;

        for (int k0 = 0; k0 < D_; k0 += 4) {
            v2f a;                                   // A = users tile (row-major)
            a.x = ua[w * 16 + m][k0 + koff];
            a.y = ua[w * 16 + m][k0 + koff + 1];
            #pragma unroll
            for (int nt = 0; nt < 3; ++nt) {         // B[k][n] = items_emb[i0+n][k]
                v2f bf;
                bf.x = ib[nt * 16 + m][k0 + koff];
                bf.y = ib[nt * 16 + m][k0 + koff + 1];
                c[nt] = __builtin_amdgcn_wmma_f32_16x16x4_f32(
                            false, a, false, bf, (short)0, c[nt], false, false);
            }
        }

        #pragma unroll
        for (int nt = 0; nt < 3; ++nt) {
            #pragma unroll
            for (int j = 0; j < 8; ++j) {
                const int mm = j + ((lane < 16) ? 0 : 8);
                const int gi = i0 + nt * 16 + m;
                if (gi < NI_)
                    out[(size_t)(u0 + w * 16 + mm) * NI_ + gi] = c[nt][j];
            }
        }
        __syncthreads();    // all waves done reading ib[it&1]

#ifdef HAVE_TDM
        if (t < 32) {
            if (it + 2 < IT_OUT) {
                const int r0 = ibase + (it + 2) * TI_OUT;
                unsigned rv = (r0 < NI_) ? (unsigned)(NI_ - r0) : 0u;
                tdm_load_rows(items_emb + (size_t)r0 * E_,
                              UA_BYTES + (unsigned)(it & 1) * IB_BYTES,
                              TI_OUT, rv, D_, 2);
                __builtin_amdgcn_s_wait_tensorcnt(1);  // next tile ready
            } else if (it + 1 < IT_OUT) {
                __builtin_amdgcn_s_wait_tensorcnt(0);  // drain last prefetch
            }
        }
        __syncthreads();
#endif
    }
}

// ---------------------------------------------------------------------
// Host-side orchestration
// ---------------------------------------------------------------------
extern "C" void kernel_launch(void* const* d_in, const int* in_sizes, int n_in,
                              void* d_out, int out_size, void* d_ws, size_t ws_size,
                              hipStream_t stream) {
    (void)in_sizes; (void)n_in; (void)out_size; (void)ws_size;

    const int*   users_ids     = (const int*)  d_in[0];
    const int*   users_items   = (const int*)  d_in[1];
    const float* users_rewards = (const float*)d_in[2];
    /* d_in[3] = step_num (== S_, compile-time constant)                */
    const float* items_table   = (const float*)d_in[4];
    /* d_in[5] = users_memory: dead — every gathered row is overwritten */
    const float* items_memory  = (const float*)d_in[6];
    const float* Wx            = (const float*)d_in[7];
    const float* Wh            = (const float*)d_in[8];
    const float* bx            = (const float*)d_in[9];
    const float* bh            = (const float*)d_in[10];
    const float* W_u           = (const float*)d_in[11];
    const float* W_i           = (const float*)d_in[12];
    float* out = (float*)d_out;

    // workspace carve-out (256B aligned), total ~104 MB
    char* ws = (char*)d_ws;
    size_t off = 0;
    auto alloc_f = [&](size_t nfloats) -> float* {
        float* p = (float*)(ws + off);
        off += ((nfloats * sizeof(float) + 255) / 256) * 256;
        return p;
    };
    float* item_sums   = alloc_f((size_t)NI_ * D_);   // 51.2 MB
    float* item_counts = alloc_f((size_t)NI_);        // 0.4 MB
    float* hT          = alloc_f((size_t)B_ * D_);    // 0.5 MB
    float* users_emb   = alloc_f((size_t)B_ * E_);    // 0.5 MB
    float* items_emb   = alloc_f((size_t)NI_ * E_);   // 51.2 MB

    const long long nz = (long long)NI_ * D_ + NI_;
    k_zero<<<dim3((unsigned)((nz + 255) / 256)), dim3(256), 0, stream>>>(
        item_sums, item_counts);

    const size_t gru_smem = (8 * X_ + 8 * D_) * sizeof(float) + 8 * sizeof(int);
    k_gru<<<dim3(B_ / 8), dim3(128), gru_smem, stream>>>(
        users_items, users_rewards, items_table, Wx, Wh, bx, bh,
        hT, item_sums, item_counts);

    k_users_emb<<<dim3(B_), dim3(128), 0, stream>>>(
        users_ids, hT, W_u, users_emb);

    const size_t ie_smem = (size_t)(D_ * E_ + 64 * (D_ + 2)) * sizeof(float); // ~97KB
    k_items_emb<<<dim3((NI_ + 63) / 64), dim3(128), ie_smem, stream>>>(
        items_memory, items_table, item_sums, item_counts, W_i, items_emb);

    const size_t out_smem = (size_t)(64 + 2 * TI_OUT) * (D_ + 2) * sizeof(float); // ~83KB
    k_out<<<dim3((NI_ + TI_OUT * IT_OUT - 1) / (TI_OUT * IT_OUT), B_ / 64),
            dim3(128), out_smem, stream>>>(users_emb, items_emb, out);
}